// Pocket_Design_49495203119125
// MI455X (gfx1250) — compile-verified
//
#include <hip/hip_runtime.h>
#include <hip/hip_bf16.h>
#include <stdint.h>

// Problem constants from the reference: B=512 proteins, R=512 residues, A=14 atoms.
#define RR   512
#define AA   14
#define NFLT (RR * AA * 3)   // 21504 floats per protein (86016 B, 16B-aligned slices)
#define NV4  (NFLT / 4)      // 5376 float4 per protein
#define BBF  (RR * 3)        // 1536 backbone floats per protein (6 KB)
#define BBV4 (BBF / 4)       // 384 B128 transfers per protein
#define NTHREADS 256

typedef float float4v __attribute__((ext_vector_type(4)));
// Types matching the async-to-LDS builtin signature: v4i addrspace(1)* src, v4i addrspace(3)* dst
typedef int v4i __attribute__((__vector_size__(16)));
typedef __attribute__((address_space(1))) v4i* as1_v4i_ptr;
typedef __attribute__((address_space(3))) v4i* as3_v4i_ptr;

static __device__ __forceinline__ int imin(int a, int b) { return a < b ? a : b; }
static __device__ __forceinline__ int imax(int a, int b) { return a > b ? a : b; }

__global__ __launch_bounds__(NTHREADS)
void pocket_interp_kernel(const float* __restrict__ predX,
                          const float* __restrict__ bb,
                          const int*   __restrict__ mask,
                          float* __restrict__ out)
{
    __shared__ float          lds_bb[BBF];    // staged backbone (async DMA target)
    __shared__ float          lds_itp[BBF];   // interpolated positions
    __shared__ int            s_prev[RR];
    __shared__ int            s_nxt[RR];
    __shared__ unsigned char  s_use[RR];
    __shared__ int            s_count;
    __shared__ int            sh_f0, sh_f1, sh_e0, sh_e1;

    const int b   = blockIdx.x;
    const int tid = threadIdx.x;

    // ---------------------------------------------------------------
    // Phase 0: launch async global->LDS copy of this protein's backbone
    // (CDNA5 async-memory path, tracked with ASYNCcnt) so it overlaps
    // with the mask scan below.
    // ---------------------------------------------------------------
    const float* gbb = bb + (size_t)b * BBF;
#if __has_builtin(__builtin_amdgcn_global_load_async_to_lds_b128)
    for (int i = tid; i < BBV4; i += NTHREADS) {
        __builtin_amdgcn_global_load_async_to_lds_b128(
            (as1_v4i_ptr)(gbb + i * 4),
            (as3_v4i_ptr)(&lds_bb[i * 4]),
            /*offset=*/0, /*cpol=*/0);
    }
#else
    for (int i = tid; i < BBV4; i += NTHREADS) {
        unsigned ldsoff = (unsigned)(uintptr_t)(&lds_bb[i * 4]);
        const float* g  = gbb + i * 4;
        asm volatile("global_load_async_to_lds_b128 %0, %1, off"
                     :: "v"(ldsoff), "v"(g) : "memory");
    }
#endif

    // ---------------------------------------------------------------
    // Phase 1: load mask, seed scans.  unm = !mask.
    //   prev seed: unm ? k : 0      (forward inclusive max-scan)
    //   nxt  seed: unm ? k : R      (backward inclusive min-scan)
    // ---------------------------------------------------------------
    const int* gmask = mask + (size_t)b * RR;
    const int k0 = tid, k1 = tid + NTHREADS;
    const int m0 = (gmask[k0] != 0);
    const int m1 = (gmask[k1] != 0);
    if (tid == 0) s_count = 0;
    s_prev[k0] = m0 ? 0  : k0;
    s_prev[k1] = m1 ? 0  : k1;
    s_nxt[k0]  = m0 ? RR : k0;
    s_nxt[k1]  = m1 ? RR : k1;
    s_use[k0]  = (unsigned char)m0;
    s_use[k1]  = (unsigned char)m1;
    __syncthreads();
    atomicAdd(&s_count, (1 - m0) + (1 - m1));

    // Hillis-Steele scans over 512 entries, 2 per thread, 9 steps.
    for (int d = 1; d < RR; d <<= 1) {
        const int p0 = s_prev[k0], p1 = s_prev[k1];
        const int a0 = (k0 >= d) ? s_prev[k0 - d] : 0;
        const int a1 = (k1 >= d) ? s_prev[k1 - d] : 0;
        const int n0 = s_nxt[k0],  n1 = s_nxt[k1];
        const int c0 = (k0 + d < RR) ? s_nxt[k0 + d] : RR;
        const int c1 = (k1 + d < RR) ? s_nxt[k1 + d] : RR;
        __syncthreads();
        s_prev[k0] = imax(p0, a0);  s_prev[k1] = imax(p1, a1);
        s_nxt[k0]  = imin(n0, c0);  s_nxt[k1]  = imin(n1, c1);
        __syncthreads();
    }

    if (tid == 0) {
        const int f0 = s_nxt[0];                       // first unmasked (or R)
        sh_f0 = f0;
        sh_f1 = (f0 + 1 < RR) ? s_nxt[f0 + 1] : RR;    // second unmasked
        const int e1 = s_prev[RR - 1];                 // last unmasked (0 if none; skip covers)
        sh_e1 = e1;
        sh_e0 = (e1 >= 1) ? s_prev[e1 - 1] : -1;       // second-to-last unmasked
    }

    // Backbone DMA must have landed in LDS before anyone reads it.
#if __has_builtin(__builtin_amdgcn_s_wait_asynccnt)
    __builtin_amdgcn_s_wait_asynccnt(0);
#else
    asm volatile("s_wait_asynccnt 0" ::: "memory");
#endif
    __syncthreads();

    // ---------------------------------------------------------------
    // Phase 2: per-residue interpolation into LDS.
    // ---------------------------------------------------------------
    const bool skip = (s_count <= 2);
    const int  f0 = sh_f0, f1 = sh_f1, e0 = sh_e0, e1 = sh_e1;
    const int  f0c = imin(f0, RR - 1), f1c = imin(f1, RR - 1);
    const int  e0c = imax(0, imin(e0, RR - 1)), e1c = imax(0, imin(e1, RR - 1));
    const float ff0 = (float)f0, ff1 = (float)f1;
    const float fe0 = (float)e0, fe1 = (float)e1;
    const float dfr = (f0 == f1) ? 1.0f : (ff0 - ff1);
    const float den = (e1 == e0) ? 1.0f : (fe1 - fe0);

#pragma unroll
    for (int t = 0; t < 2; ++t) {
        const int k  = tid + t * NTHREADS;
        const int pk = s_prev[k];
        const int nk = s_nxt[k];
        const int pc = pk;                 // already in [0, R-1]
        const int nc = imin(nk, RR - 1);
        const float kf = (float)k;
        float dmid = (float)(nk - pk);
        dmid = (dmid == 0.0f) ? 1.0f : dmid;
        const float wf = (kf - ff0) / dfr;
        const float we = (kf - fe1) / den;
        const float wp = kf - (float)pk;
        const float wn = (float)nk - kf;
#pragma unroll
        for (int c = 0; c < 3; ++c) {
            const float bf0 = lds_bb[f0c * 3 + c];
            const float bf1 = lds_bb[f1c * 3 + c];
            const float be0 = lds_bb[e0c * 3 + c];
            const float be1 = lds_bb[e1c * 3 + c];
            const float bp  = lds_bb[pc  * 3 + c];
            const float bn  = lds_bb[nc  * 3 + c];
            const float fr  = bf0 + wf * (bf0 - bf1);          // front extrapolation
            const float en  = be1 + we * (be1 - be0);          // end extrapolation
            const float md  = (wp * bn + wn * bp) / dmid;      // mid interpolation
            lds_itp[k * 3 + c] = (k < f0) ? fr : ((k > e1) ? en : md);
        }
        s_use[k] = (s_use[k] && !skip) ? (unsigned char)1 : (unsigned char)0;
    }
    __syncthreads();

    // ---------------------------------------------------------------
    // Phase 3: stream pred_X -> out with per-element select (B128 I/O).
    // Per protein slice is 86016 B = 16B-aligned; chunks never misalign.
    // flat = r*42 + a*3 + c  ->  r = flat/42, c = (flat % 42) % 3.
    // ---------------------------------------------------------------
    const float4v* __restrict__ src4 = (const float4v*)(predX + (size_t)b * NFLT);
    float4v*       __restrict__ dst4 = (float4v*)(out   + (size_t)b * NFLT);
    for (int i = tid; i < NV4; i += NTHREADS) {
        float4v v = src4[i];
        const int base = i * 4;
#pragma unroll
        for (int j = 0; j < 4; ++j) {
            const int flat = base + j;
            const int r = flat / 42;
            const int c = (flat - r * 42) % 3;
            if (s_use[r]) v[j] = lds_itp[r * 3 + c];
        }
        dst4[i] = v;
    }
}

extern "C" void kernel_launch(void* const* d_in, const int* in_sizes, int n_in,
                              void* d_out, int out_size, void* d_ws, size_t ws_size,
                              hipStream_t stream) {
    const float* predX = (const float*)d_in[0];   // [B, R, 14, 3] f32
    const float* bbpos = (const float*)d_in[1];   // [B, R, 3]    f32
    const int*   rmask = (const int*)  d_in[2];   // [B, R]       bool->int
    float*       out   = (float*)d_out;           // [B, R, 14, 3] f32

    const int B = in_sizes[2] / RR;               // 512
    pocket_interp_kernel<<<B, NTHREADS, 0, stream>>>(predX, bbpos, rmask, out);
}